// GRU_59219009077899
// MI455X (gfx1250) — compile-verified
//
#include <hip/hip_runtime.h>
#include <hip/hip_bf16.h>

// ---------------------------------------------------------------------------
// GRU cell + tanh readout for MI455X (gfx1250, wave32).
//
// Roofline: dominant op is logit = tanh(h1 @ w_out.T + b_out):
//   13.1 GFLOP f32, 51.2 MB read + 102.4 MB write -> HBM-bound (~6.6us @23.3TB/s).
//   => keep f32 precision and use V_WMMA_F32_16X16X4_F32 (no bf16 downcast
//      needed; compute is free relative to memory).
// h1 (0.5MB) and w_out (51.2MB) are L2-resident (192MB), so a simple
// 16x128-per-workgroup tiling reaches the HBM bound via L2 reuse.
// ---------------------------------------------------------------------------

typedef __attribute__((ext_vector_type(2))) float v2f;
typedef __attribute__((ext_vector_type(8))) float v8f;

#define KDIM 256          // inner dim for both GEMMs (H = 256)
#define LDS_STRIDE 260    // 256 + 4 pad: A-reads hit 16 distinct banks

// tanh via native v_exp_f32 (no ftanh libcall on this toolchain).
// Overflow-safe: exp(2|x|) -> inf => 2/inf -> 0 => result -> +/-1.
__device__ __forceinline__ float tanh_exp(float x) {
  const float ax = __builtin_fabsf(x);
  const float e  = __builtin_expf(2.0f * ax);
  const float t  = 1.0f - 2.0f / (e + 1.0f);
  return __builtin_copysignf(t, x);
}

__device__ __forceinline__ float sigmoid_exp(float x) {
  return 1.0f / (1.0f + __builtin_expf(-x));
}

// C[m,n] = act( sum_k A[m,k] * W[n,k] + bias[n] )
// A: M x 256 row-major, W: N x 256 row-major (so this is A @ W^T), C: M x N.
// Workgroup: 256 threads = 8 waves; covers 16 rows x 128 cols.
// Each wave: one 16x16 tile via 64 x v_wmma_f32_16x16x4_f32.
template <bool TANH>
__global__ __launch_bounds__(256) void gemm_bias_act_wmma(
    const float* __restrict__ A, const float* __restrict__ W,
    const float* __restrict__ bias, float* __restrict__ C, int N) {
  __shared__ float As[16 * LDS_STRIDE];

  const int row0 = blockIdx.y * 16;
  const int tid  = threadIdx.x;

  // Stage A tile (16 x 256) into LDS, coalesced: iteration i loads row i,
  // thread tid loads column tid (1KB contiguous per row).
  #pragma unroll
  for (int i = 0; i < 16; ++i)
    As[i * LDS_STRIDE + tid] = A[(size_t)(row0 + i) * KDIM + tid];
  __syncthreads();

  const int wave = tid >> 5;
  const int lane = tid & 31;
  const int col0 = blockIdx.x * 128 + wave * 16;
  if (col0 >= N) return;           // uniform per-wave (N % 16 == 0), EXEC stays all-1s

  const int half = lane >> 4;      // 0: lanes 0-15, 1: lanes 16-31
  const int l    = lane & 15;

  // B operand source: row (col0+l) of W; lane half selects K offset {0,2}.
  const float* wrow = W + (size_t)(col0 + l) * KDIM;

  v8f acc = {};
  #pragma unroll 8
  for (int k0 = 0; k0 < KDIM; k0 += 4) {
    const int kk = k0 + 2 * half;
    // A 16x4 f32 layout: lanes 0-15 hold (M=lane, K={k0,k0+1}),
    //                    lanes 16-31 hold (M=lane-16, K={k0+2,k0+3}).
    v2f a;
    a.x = As[l * LDS_STRIDE + kk];
    a.y = As[l * LDS_STRIDE + kk + 1];
    // B 4x16 f32 layout (symmetric): lane half picks K {+0,+2}, lane = N.
    // 8-byte aligned: row base is 1KB-aligned, kk is even.
    v2f b = *(const v2f*)(wrow + kk);
    // 8 args: (neg_a, A, neg_b, B, c_mod, C, reuse_a, reuse_b)
    acc = __builtin_amdgcn_wmma_f32_16x16x4_f32(
        false, a, false, b, (short)0, acc, false, false);
  }

  // C/D layout: VGPR r -> row (r + 8*half), col = lane&15.
  const int col = col0 + l;
  const float bv = bias[col];
  #pragma unroll
  for (int r = 0; r < 8; ++r) {
    const int row = row0 + r + 8 * half;
    float v = acc[r] + bv;
    if (TANH) v = tanh_exp(v);
    C[(size_t)row * N + col] = v;
  }
}

// Fused gather + gates + hidden-state blend.
// x_proj[b,j] = w_ih[j, input[b]] + b_ih[j]   (column gather, stride V)
// h_proj already contains +b_hh (folded into the GEMM bias).
__global__ __launch_bounds__(256) void gru_gates(
    const int* __restrict__ input, const float* __restrict__ h0,
    const float* __restrict__ w_ih, const float* __restrict__ b_ih,
    const float* __restrict__ hproj, float* __restrict__ h1_ws,
    float* __restrict__ h1_out) {
  const int gid = blockIdx.x * 256 + threadIdx.x;  // B*H = 131072 threads
  const int b = gid >> 8;                          // H == 256
  const int j = gid & 255;
  const size_t V = 50000;
  const int idx = input[b];

  const float xr = w_ih[(size_t)j * V + idx] + b_ih[j];
  const float xz = w_ih[(size_t)(j + 256) * V + idx] + b_ih[j + 256];
  const float xn = w_ih[(size_t)(j + 512) * V + idx] + b_ih[j + 512];

  const float hr = hproj[b * 768 + j];
  const float hz = hproj[b * 768 + j + 256];
  const float hn = hproj[b * 768 + j + 512];

  const float r = sigmoid_exp(xr + hr);
  const float z = sigmoid_exp(xz + hz);
  const float n = tanh_exp(xn + r * hn);

  const float h = (1.0f - z) * n + z * h0[gid];
  h1_ws[gid] = h;
  h1_out[gid] = h;
}

extern "C" void kernel_launch(void* const* d_in, const int* in_sizes, int n_in,
                              void* d_out, int out_size, void* d_ws, size_t ws_size,
                              hipStream_t stream) {
  (void)in_sizes; (void)n_in; (void)out_size; (void)ws_size;
  const int*   input  = (const int*)d_in[0];
  // d_in[1] = target (unused in forward)
  const float* hidden = (const float*)d_in[2];   // (1, 512, 256)
  const float* w_ih   = (const float*)d_in[3];   // (768, 50000)
  const float* w_hh   = (const float*)d_in[4];   // (768, 256)
  const float* b_ih   = (const float*)d_in[5];   // (768,)
  const float* b_hh   = (const float*)d_in[6];   // (768,)
  const float* w_out  = (const float*)d_in[7];   // (50000, 256)
  const float* b_out  = (const float*)d_in[8];   // (50000,)

  float* logit  = (float*)d_out;                        // 512*50000
  float* h1_out = logit + (size_t)512 * 50000;          // 512*256

  float* hproj = (float*)d_ws;                          // 512*768 f32
  float* h1_ws = hproj + 512 * 768;                     // 512*256 f32

  // 1) h_proj = h0 @ w_hh^T + b_hh      (M=512, N=768, K=256)
  dim3 g1(768 / 128, 512 / 16);
  gemm_bias_act_wmma<false><<<g1, 256, 0, stream>>>(hidden, w_hh, b_hh, hproj, 768);

  // 2) gates: gather x_proj, sigmoid/tanh, blend -> h1
  gru_gates<<<(512 * 256) / 256, 256, 0, stream>>>(
      input, hidden, w_ih, b_ih, hproj, h1_ws, h1_out);

  // 3) logit = tanh(h1 @ w_out^T + b_out)   (M=512, N=50000, K=256)
  dim3 g3((50000 + 127) / 128, 512 / 16);
  gemm_bias_act_wmma<true><<<g3, 256, 0, stream>>>(h1_ws, w_out, b_out, logit, 50000);
}